// ModelRollout_12833362280837
// MI455X (gfx1250) — compile-verified
//
#include <hip/hip_runtime.h>
#include <stdint.h>

// ---------------------------------------------------------------------------
// Tsit5 fixed-step rollout of a 5-state nonlinear ODE, 2M sequential steps.
// Serial-latency-bound: one wave owns the dependent chain; output staging is
// pushed off the critical path via LDS double-buffering + CDNA5 async
// LDS->global DMA stores (global_store_async_from_lds_b128, ASYNCcnt).
// ---------------------------------------------------------------------------

#define CHUNK 128   // steps staged per LDS buffer; 128 floats/row = 512B = one b128 async store per row
#define NROWS 10    // 5 y rows + 4 w rows + 1 t row

// Tsit5 tableau (f32, matching the f32 reference constants)
__constant__ static const float cA21 = 0.161f;
__constant__ static const float cA31 = -0.008480655492356989f, cA32 = 0.335480655492357f;
__constant__ static const float cA41 = 2.8971530571054935f,  cA42 = -6.359448489975075f,  cA43 = 4.3622954328695815f;
__constant__ static const float cA51 = 5.325864828439257f,   cA52 = -11.748883564062828f, cA53 = 7.4955393428898365f, cA54 = -0.09249506636175525f;
__constant__ static const float cA61 = 5.86145544294642f,    cA62 = -12.92096931784711f,  cA63 = 8.159367898576159f,  cA64 = -0.071584973281401f, cA65 = -0.028269050394068383f;
__constant__ static const float cB1  = 0.09646076681806523f, cB2  = 0.01f,                cB3  = 0.4798896504144996f,
                                cB4  = 1.379008574103742f,   cB5  = -3.290069515436081f,  cB6  = 2.324710524099774f;

// Vector field: recompute algebraic assignments w from y, then the rate rules.
__device__ __forceinline__ void vf(const float y[5], const float C[13], float d[5]) {
    const float w0 = C[1] - y[2];       // Effectorina
    const float w1 = y[3] + y[4];       // Damage
    const float w2 = C[2] - y[0];       // Timerinact
    const float w3 = C[0] - y[1];       // CellCina
    d[0] = C[3] * w1 * w2 / (C[10] + w2) - C[4] * y[0] / (C[10] + y[0]);
    d[1] = (C[7] + y[1]) * w3 / (C[11] + w3)
         - C[6] * y[0] * y[1] / (C[11] + y[1])
         - C[9] * y[1] * y[2] / (C[11] + y[1]);
    d[2] = C[8] * w1 * w0 / (C[11] + w0) - C[5] * y[1] * y[2] / (C[11] + y[2]);
    d[3] = -0.2f * y[3];
    d[4] = -0.5f * y[4];
}

__device__ __forceinline__ void tsit5_step(float y[5], const float C[13], float dt) {
    float k1[5], k2[5], k3[5], k4[5], k5[5], k6[5], yt[5];
    vf(y, C, k1);
#pragma unroll
    for (int i = 0; i < 5; ++i) yt[i] = y[i] + dt * (cA21 * k1[i]);
    vf(yt, C, k2);
#pragma unroll
    for (int i = 0; i < 5; ++i) yt[i] = y[i] + dt * (cA31 * k1[i] + cA32 * k2[i]);
    vf(yt, C, k3);
#pragma unroll
    for (int i = 0; i < 5; ++i) yt[i] = y[i] + dt * (cA41 * k1[i] + cA42 * k2[i] + cA43 * k3[i]);
    vf(yt, C, k4);
#pragma unroll
    for (int i = 0; i < 5; ++i) yt[i] = y[i] + dt * (cA51 * k1[i] + cA52 * k2[i] + cA53 * k3[i] + cA54 * k4[i]);
    vf(yt, C, k5);
#pragma unroll
    for (int i = 0; i < 5; ++i) yt[i] = y[i] + dt * (cA61 * k1[i] + cA62 * k2[i] + cA63 * k3[i] + cA64 * k4[i] + cA65 * k5[i]);
    vf(yt, C, k6);
#pragma unroll
    for (int i = 0; i < 5; ++i)
        y[i] = y[i] + dt * (cB1 * k1[i] + cB2 * k2[i] + cB3 * k3[i] + cB4 * k4[i] + cB5 * k5[i] + cB6 * k6[i]);
}

__global__ __launch_bounds__(32)
void tsit5_rollout_kernel(const float* __restrict__ y0p,
                          const float* __restrict__ w0p,
                          const float* __restrict__ cp,
                          float* __restrict__ out,
                          long long n) // n_steps; out is [10, n] row-major
{
    // Double-buffered staging: [buffer][row][column]. 2*10*128*4 = 10 KB of 320 KB LDS.
    __shared__ float buf[2][NROWS][CHUNK];

    float C[13];
#pragma unroll
    for (int i = 0; i < 13; ++i) C[i] = cp[i];
    float y[5];
#pragma unroll
    for (int i = 0; i < 5; ++i) y[i] = y0p[i];
    float w[4];
#pragma unroll
    for (int i = 0; i < 4; ++i) w[i] = w0p[i]; // step 0 emits the *input* w0 (scan emits pre-step state)

    float t = 0.0f;
    const float dt = 0.1f;
    const int lane = (int)(threadIdx.x & 31u);
    int b = 0;

    for (long long s = 0; s < n; s += CHUNK) {
        const int len = ((n - s) < (long long)CHUNK) ? (int)(n - s) : CHUNK;

        // Buffer b was last drained by 10 async stores issued one buffer-cycle ago.
        // Async stores complete in order, so ASYNCcnt <= 10 guarantees buffer b is free
        // (at most the other buffer's 10 stores remain outstanding).
        asm volatile("s_wait_asynccnt 10" ::: "memory");

        for (int col = 0; col < len; ++col) {
            if (lane == 0) { // stage pre-step (y, w, t); DS pipe, off the VALU critical path
                buf[b][0][col] = y[0];
                buf[b][1][col] = y[1];
                buf[b][2][col] = y[2];
                buf[b][3][col] = y[3];
                buf[b][4][col] = y[4];
                buf[b][5][col] = w[0];
                buf[b][6][col] = w[1];
                buf[b][7][col] = w[2];
                buf[b][8][col] = w[3];
                buf[b][9][col] = t;
            }
            tsit5_step(y, C, dt);
            // w_new = assign(y_new)
            w[0] = C[1] - y[2];
            w[1] = y[3] + y[4];
            w[2] = C[2] - y[0];
            w[3] = C[0] - y[1];
            t += dt; // f32 accumulation, matching the scan carry
        }

        if (len == CHUNK) {
            // Make the wave's DS writes visible to the async DMA engine, then fire
            // one async b128 store per row: 32 lanes x 16B = 512B contiguous burst.
            asm volatile("s_wait_dscnt 0" ::: "memory");
#pragma unroll
            for (int r = 0; r < NROWS; ++r) {
                unsigned long long ga =
                    (unsigned long long)(uintptr_t)(out + (size_t)r * (size_t)n + (size_t)s)
                    + (unsigned long long)(lane * 16);
                unsigned int la =
                    (unsigned int)(uintptr_t)(&buf[b][r][0]) + (unsigned int)(lane * 16);
                asm volatile("global_store_async_from_lds_b128 %0, %1, off"
                             :: "v"(ga), "v"(la)
                             : "memory");
            }
        } else {
            // Partial tail chunk: plain coalesced stores (robust to any n_steps).
            __syncthreads();
            for (int r = 0; r < NROWS; ++r)
                for (int col = lane; col < len; col += 32)
                    out[(size_t)r * (size_t)n + (size_t)s + (size_t)col] = buf[b][r][col];
            __syncthreads();
        }
        b ^= 1;
    }
    // s_endpgm performs an implicit wait-idle, draining outstanding async stores.
}

extern "C" void kernel_launch(void* const* d_in, const int* in_sizes, int n_in,
                              void* d_out, int out_size, void* d_ws, size_t ws_size,
                              hipStream_t stream) {
    (void)in_sizes; (void)n_in; (void)d_ws; (void)ws_size;
    // setup_inputs order: t1 (int, unused: n derivable from out_size), y0[5], w0[4], c[13]
    const float* y0 = (const float*)d_in[1];
    const float* w0 = (const float*)d_in[2];
    const float* c  = (const float*)d_in[3];
    float* out = (float*)d_out;

    const long long n_steps = (long long)out_size / 10; // out = [5+4+1, n_steps] f32, flat

    // One wave32 owns the serial recurrence; the rest of the chip cannot help
    // (strict step-to-step dependence). Async LDS->global stores keep the
    // 80 MB of trajectory output entirely off the dependent chain.
    tsit5_rollout_kernel<<<1, 32, 0, stream>>>(y0, w0, c, out, n_steps);
}